// AdaGNN_62981400428667
// MI455X (gfx1250) — compile-verified
//
#include <hip/hip_runtime.h>
#include <hip/hip_bf16.h>

typedef __attribute__((ext_vector_type(2))) float v2f;
typedef __attribute__((ext_vector_type(8))) float v8f;

#define HDIM 64

// ---------------------------------------------------------------------------
// GEMM: Out[n, 0:64] (+)= relu?( alpha * (A[n,:K] .* ascale) @ B^T + bias )
//   A: (nrows, lda) row-major, B: (64, ldb) row-major (weights, out = A @ B^T)
// One wave computes a 16x64 block with V_WMMA_F32_16X16X4_F32.
// K / scale / relu / accum are compile-time so the K-loop fully unrolls into
// immediate-offset b64 loads overlapped with the WMMA chain.
// ---------------------------------------------------------------------------
template <int K, bool HAS_SCALE, bool RELU, bool ACCUM>
__global__ __launch_bounds__(256) void k_gemm_wmma(
    const float* __restrict__ A, int lda,
    const float* __restrict__ B, int ldb,
    const float* __restrict__ ascale,   // per-k scale (len K); used iff HAS_SCALE
    float alpha,                        // used iff HAS_SCALE
    const float* __restrict__ bias,     // optional (len 64) or null
    float* __restrict__ Out,
    int nrows)
{
  const int wavesPerBlock = blockDim.x >> 5;
  const int wave = blockIdx.x * wavesPerBlock + (threadIdx.x >> 5);
  const int row0 = wave << 4;
  if (row0 >= nrows) return;                 // whole-wave uniform (EXEC all 1s)
  const int lane = threadIdx.x & 31;
  const int lid  = lane & 15;
  const int half = lane >> 4;

  v8f acc[4] = {};

  const float* Arow = A + (size_t)(row0 + lid) * lda + half * 2;
  const float* Brow0 = B + (size_t)(0 * 16 + lid) * ldb + half * 2;
  const float* Brow1 = B + (size_t)(1 * 16 + lid) * ldb + half * 2;
  const float* Brow2 = B + (size_t)(2 * 16 + lid) * ldb + half * 2;
  const float* Brow3 = B + (size_t)(3 * 16 + lid) * ldb + half * 2;
  const float* Srow  = HAS_SCALE ? (ascale + half * 2) : nullptr;

#pragma unroll
  for (int k0 = 0; k0 < K; k0 += 4) {
    v2f av = *(const v2f*)(Arow + k0);
    if (HAS_SCALE) {
      v2f sv = *(const v2f*)(Srow + k0);
      av[0] *= alpha * sv[0];
      av[1] *= alpha * sv[1];
    }
    v2f b0 = *(const v2f*)(Brow0 + k0);
    v2f b1 = *(const v2f*)(Brow1 + k0);
    v2f b2 = *(const v2f*)(Brow2 + k0);
    v2f b3 = *(const v2f*)(Brow3 + k0);
    acc[0] = __builtin_amdgcn_wmma_f32_16x16x4_f32(false, av, false, b0, (short)0, acc[0], false, false);
    acc[1] = __builtin_amdgcn_wmma_f32_16x16x4_f32(false, av, false, b1, (short)0, acc[1], false, false);
    acc[2] = __builtin_amdgcn_wmma_f32_16x16x4_f32(false, av, false, b2, (short)0, acc[2], false, false);
    acc[3] = __builtin_amdgcn_wmma_f32_16x16x4_f32(false, av, false, b3, (short)0, acc[3], false, false);
  }

#pragma unroll
  for (int ct = 0; ct < 4; ++ct) {
    const int col = ct * 16 + lid;
    const float bv = bias ? bias[col] : 0.0f;
#pragma unroll
    for (int r = 0; r < 8; ++r) {
      const int row = row0 + r + half * 8;   // C/D layout: halves hold M / M+8
      float v = acc[ct][r] + bv;
      if (RELU) v = fmaxf(v, 0.0f);
      float* p = Out + (size_t)row * HDIM + col;
      if (ACCUM) v += *p;
      *p = v;
    }
  }
}

// ---------------------------------------------------------------------------
__global__ void k_fill(float* __restrict__ p, long long n, float v) {
  long long t = (long long)blockIdx.x * blockDim.x + threadIdx.x;
  if (t < n) p[t] = v;
}

__global__ void k_edge_deg(const int* __restrict__ dst, float* __restrict__ deg, int E) {
  int e = blockIdx.x * blockDim.x + threadIdx.x;
  if (e < E) atomicAdd(&deg[dst[e]], 1.0f);
}

__global__ void k_dinv(float* __restrict__ deg, int n) {
  int i = blockIdx.x * blockDim.x + threadIdx.x;
  if (i < n) deg[i] = 1.0f / sqrtf(fmaxf(deg[i], 1.0f));
}

// agg[dst] += feat[src] * dinv[src]; 16 threads per edge, float4 each.
__global__ void k_edge_scatter(const int* __restrict__ src, const int* __restrict__ dst,
                               const float* __restrict__ feat, const float* __restrict__ dinv,
                               float* __restrict__ agg, int E)
{
  long long t = (long long)blockIdx.x * blockDim.x + threadIdx.x;
  int e = (int)(t >> 4);
  if (e >= E) return;
  int c = ((int)t & 15) << 2;
  int s = src[e];
  int d = dst[e];
  float sc = dinv[s];
  const float4 f = *(const float4*)(feat + (size_t)s * HDIM + c);
  float* ap = agg + (size_t)d * HDIM + c;
  atomicAdd(ap + 0, f.x * sc);
  atomicAdd(ap + 1, f.y * sc);
  atomicAdd(ap + 2, f.z * sc);
  atomicAdd(ap + 3, f.w * sc);
}

// v = fin - agg*dinv*ld ; (optional) fout = v ; hbr += theta * v
__global__ void k_feat_update(const float* __restrict__ fin, const float* __restrict__ agg,
                              const float* __restrict__ dinv, const float* __restrict__ ld,
                              float theta, float* __restrict__ fout, float* __restrict__ hbr,
                              int n)
{
  long long t = (long long)blockIdx.x * blockDim.x + threadIdx.x;
  int node = (int)(t >> 4);
  if (node >= n) return;
  int c = ((int)t & 15) << 2;
  size_t off = (size_t)node * HDIM + c;
  float di = dinv[node];
  float4 fi = *(const float4*)(fin + off);
  float4 ag = *(const float4*)(agg + off);
  float4 l  = *(const float4*)(ld + c);
  float4 v;
  v.x = fi.x - ag.x * di * l.x;
  v.y = fi.y - ag.y * di * l.y;
  v.z = fi.z - ag.z * di * l.z;
  v.w = fi.w - ag.w * di * l.w;
  if (fout) *(float4*)(fout + off) = v;
  float4 h = *(const float4*)(hbr + off);
  h.x += theta * v.x;
  h.y += theta * v.y;
  h.z += theta * v.z;
  h.w += theta * v.w;
  *(float4*)(hbr + off) = h;
}

// emb = relu(embpre + b3)  (written to output); logits = emb @ W4^T + b4
__global__ void k_final(const float* __restrict__ embpre, const float* __restrict__ b3,
                        const float* __restrict__ W4, const float* __restrict__ b4,
                        float* __restrict__ logits, float* __restrict__ emb, int n)
{
  int node = blockIdx.x * blockDim.x + threadIdx.x;
  if (node >= n) return;
  const float* rp = embpre + (size_t)node * HDIM;
  float* ep = emb + (size_t)node * HDIM;
  float a0 = b4[0], a1 = b4[1];
#pragma unroll
  for (int k = 0; k < HDIM; k += 4) {
    float4 v  = *(const float4*)(rp + k);
    float4 bb = *(const float4*)(b3 + k);
    float4 w0 = *(const float4*)(W4 + k);
    float4 w1 = *(const float4*)(W4 + HDIM + k);
    v.x = fmaxf(v.x + bb.x, 0.0f);
    v.y = fmaxf(v.y + bb.y, 0.0f);
    v.z = fmaxf(v.z + bb.z, 0.0f);
    v.w = fmaxf(v.w + bb.w, 0.0f);
    *(float4*)(ep + k) = v;
    a0 += v.x * w0.x + v.y * w0.y + v.z * w0.z + v.w * w0.w;
    a1 += v.x * w1.x + v.y * w1.y + v.z * w1.z + v.w * w1.w;
  }
  logits[(size_t)node * 2 + 0] = a0;
  logits[(size_t)node * 2 + 1] = a1;
}

// ---------------------------------------------------------------------------
extern "C" void kernel_launch(void* const* d_in, const int* in_sizes, int n_in,
                              void* d_out, int out_size, void* d_ws, size_t ws_size,
                              hipStream_t stream)
{
  (void)n_in; (void)out_size; (void)ws_size;
  const float* in_feat = (const float*)d_in[0];
  const int*   src     = (const int*)d_in[1];
  const int*   dst     = (const int*)d_in[2];
  const float* W1 = (const float*)d_in[3];
  const float* b1 = (const float*)d_in[4];
  const float* W2 = (const float*)d_in[5];
  const float* b2 = (const float*)d_in[6];
  const float* W3 = (const float*)d_in[7];   // (64, 192) row-major
  const float* b3 = (const float*)d_in[8];
  const float* W4 = (const float*)d_in[9];   // (2, 64) row-major
  const float* b4 = (const float*)d_in[10];

  const int INDIM = 128;
  const int N = in_sizes[0] / INDIM;
  const int E = in_sizes[1];
  const long long NH = (long long)N * HDIM;

  // workspace layout (fp32)
  float* ws    = (float*)d_ws;
  float* deg   = ws;                          // N  (becomes dinv in place)
  float* Bh    = deg + N;                     // N*64  h after MLP
  float* Bt    = Bh    + (size_t)NH;          // N*64  h0 temp, then feat1
  float* Bagg1 = Bt    + (size_t)NH;          // N*64  shared first propagation
  float* Bagg2 = Bagg1 + (size_t)NH;          // N*64  per-branch propagation
  float* Bhbr  = Bagg2 + (size_t)NH;          // N*64  branch output
  float* Bemb  = Bhbr  + (size_t)NH;          // N*64  pre-activation emb accumulator

  float* out    = (float*)d_out;
  float* logits = out;                        // (N,2)
  float* emb    = out + (size_t)2 * N;        // (N,64)

  const int TB = 256;
  auto cdiv = [](long long a, long long b) { return (int)((a + b - 1) / b); };

  const int tiles = cdiv(N, 16);
  const int gG = cdiv(tiles, 8);              // 8 waves / block

  // THETAS[i] = (t0, t1, t2)
  const float TH[3][3] = {{3.0f, -3.0f, 0.75f},
                          {0.0f,  3.0f, -1.5f},
                          {0.0f,  0.0f, 0.75f}};

  // degree -> dinv
  k_fill<<<cdiv(N, TB), TB, 0, stream>>>(deg, N, 0.0f);
  k_edge_deg<<<cdiv(E, TB), TB, 0, stream>>>(dst, deg, E);
  k_dinv<<<cdiv(N, TB), TB, 0, stream>>>(deg, N);

  // h = relu(relu(in_feat @ W1^T + b1) @ W2^T + b2)
  k_gemm_wmma<128, false, true, false><<<gG, 256, 0, stream>>>(
      in_feat, INDIM, W1, INDIM, nullptr, 1.0f, b1, Bt, N);
  k_gemm_wmma<64, false, true, false><<<gG, 256, 0, stream>>>(
      Bt, HDIM, W2, HDIM, nullptr, 1.0f, b2, Bh, N);

  // shared first propagation: agg1 = segsum((h*dinv)[src] -> dst)
  k_fill<<<cdiv(NH, TB), TB, 0, stream>>>(Bagg1, NH, 0.0f);
  k_edge_scatter<<<cdiv((long long)E * 16, TB), TB, 0, stream>>>(src, dst, Bh, deg, Bagg1, E);
  k_fill<<<cdiv(NH, TB), TB, 0, stream>>>(Bemb, NH, 0.0f);

  for (int i = 0; i < 3; ++i) {
    const float* ldi = (const float*)d_in[11 + 3 * i];  // (3,64)
    const float* cWi = (const float*)d_in[12 + 3 * i];  // (64,64)
    const float* cbi = (const float*)d_in[13 + 3 * i];  // (64,)
    const float t0 = TH[i][0], t1 = TH[i][1], t2 = TH[i][2];

    // hbr = t0 * (h .* ld[0]) @ cW^T + cb
    k_gemm_wmma<64, true, false, false><<<gG, 256, 0, stream>>>(
        Bh, HDIM, cWi, HDIM, ldi, t0, cbi, Bhbr, N);

    // feat1 = h - agg1*dinv*ld[1]; hbr += t1*feat1  (feat1 -> Bt)
    k_feat_update<<<cdiv((long long)N * 16, TB), TB, 0, stream>>>(
        Bh, Bagg1, deg, ldi + HDIM, t1, Bt, Bhbr, N);

    // second propagation on feat1
    k_fill<<<cdiv(NH, TB), TB, 0, stream>>>(Bagg2, NH, 0.0f);
    k_edge_scatter<<<cdiv((long long)E * 16, TB), TB, 0, stream>>>(src, dst, Bt, deg, Bagg2, E);

    // feat2 = feat1 - agg2*dinv*ld[2]; hbr += t2*feat2  (feat2 not stored)
    k_feat_update<<<cdiv((long long)N * 16, TB), TB, 0, stream>>>(
        Bt, Bagg2, deg, ldi + 2 * HDIM, t2, nullptr, Bhbr, N);

    // embpre += hbr @ W3[:, i*64:(i+1)*64]^T
    k_gemm_wmma<64, false, false, true><<<gG, 256, 0, stream>>>(
        Bhbr, HDIM, W3 + i * HDIM, 3 * HDIM, nullptr, 1.0f, nullptr, Bemb, N);
  }

  // emb = relu(embpre + b3); logits = emb @ W4^T + b4
  k_final<<<cdiv(N, TB), TB, 0, stream>>>(Bemb, b3, W4, b4, logits, emb, N);
}